// HyperConv_36833639531048
// MI455X (gfx1250) — compile-verified
//
#include <hip/hip_runtime.h>

typedef __attribute__((ext_vector_type(2))) float v2f;
typedef __attribute__((ext_vector_type(8))) float v8f;

#define IN_CH   16
#define OUT_CH  16
#define KTAP    3
#define DIL     4
#define CONDD   8
#define BATCH   32
#define TOUT    65536
#define PADX    (DIL * (KTAP - 1))   // 8
#define TIN     (TOUT + PADX)        // 65544
#define CTOT    (IN_CH * KTAP)       // 48  (K dimension of the GEMM)
#define NCHUNK  (CTOT / 4)           // 12 WMMA_F32_16X16X4 steps per tile
#define TILES_PER_WAVE 16            // 256 time positions per wave

// ---------------------------------------------------------------------------
// Kernel 1: hypernetwork.  wbuf[b][c][o] = b2 + leakyrelu(p@W1+b1) @ W2
// Tiny: 32 * 768 outputs. One block per batch sample.
// ---------------------------------------------------------------------------
__global__ __launch_bounds__(256) void hypernet_kernel(
    const float* __restrict__ p,  const float* __restrict__ W1,
    const float* __restrict__ b1, const float* __restrict__ W2,
    const float* __restrict__ b2, float* __restrict__ wbuf)
{
    const int b   = blockIdx.x;
    const int tid = threadIdx.x;
    __shared__ float h[IN_CH];

    if (tid < IN_CH) {
        float acc = b1[tid];
        #pragma unroll
        for (int i = 0; i < CONDD; ++i)
            acc = fmaf(p[b * CONDD + i], W1[i * IN_CH + tid], acc);
        h[tid] = (acc > 0.0f) ? acc : 0.2f * acc;   // LeakyReLU(0.2)
    }
    __syncthreads();

    for (int idx = tid; idx < CTOT * OUT_CH; idx += 256) {
        float acc = b2[idx];
        #pragma unroll
        for (int j = 0; j < IN_CH; ++j)
            acc = fmaf(h[j], W2[j * (CTOT * OUT_CH) + idx], acc);
        wbuf[b * (CTOT * OUT_CH) + idx] = acc;      // [48][16] row-major (c,o)
    }
}

// ---------------------------------------------------------------------------
// Kernel 2: batched skinny GEMM via V_WMMA_F32_16X16X4_F32.
// Per wave: one batch b, strip of 256 time positions = 16 output tiles of
// 16(out_ch) x 16(t).  K = 48 channel-taps -> 12 WMMAs per tile.
//
// A (16x4, M=o, K=c):  VGPR0 lanes0-15 K=0 / lanes16-31 K=2; VGPR1 K=1/K=3.
// B (4x16, K=c, N=t):  same K striping, N = lane%16.
// D (16x16):           VGPR r, lane l -> o = r + 8*(l>=16), t = t0 + l%16.
// ---------------------------------------------------------------------------
__global__ __launch_bounds__(256) void hyperconv_main(
    const float* __restrict__ x, const float* __restrict__ wbuf,
    float* __restrict__ y)
{
    const int lane  = threadIdx.x & 31;
    const int half  = lane >> 4;       // 0: lanes 0-15, 1: lanes 16-31
    const int ln    = lane & 15;
    const int gwave = blockIdx.x * (blockDim.x >> 5) + (threadIdx.x >> 5);
    const int b     = gwave >> 8;      // 256 waves per batch sample
    const int strip = gwave & 255;
    const int t0    = strip * (16 * TILES_PER_WAVE);

    // --- A operands: per-batch hyper-weights, resident for the whole strip ---
    v2f A[NCHUNK];
    {
        const float* wb = wbuf + b * (CTOT * OUT_CH);
        #pragma unroll
        for (int j = 0; j < NCHUNK; ++j) {
            const int c0 = 4 * j + 2 * half;
            A[j][0] = wb[(c0 + 0) * OUT_CH + ln];  // K = 2*half
            A[j][1] = wb[(c0 + 1) * OUT_CH + ln];  // K = 2*half + 1
        }
    }

    // --- per-lane element offsets into x for the two B rows of each chunk ---
    // x_cat[c][t] = x[b][c%16][t + PADX - DIL*(c/16)]
    int offs0[NCHUNK], offs1[NCHUNK];
    #pragma unroll
    for (int j = 0; j < NCHUNK; ++j) {
        const int c0 = 4 * j + 2 * half;
        const int ca = c0,     ta = ca >> 4, cha = ca & 15;
        const int cb = c0 + 1, tb = cb >> 4, chb = cb & 15;
        offs0[j] = (b * IN_CH + cha) * TIN + (PADX - DIL * ta) + t0 + ln;
        offs1[j] = (b * IN_CH + chb) * TIN + (PADX - DIL * tb) + t0 + ln;
    }

    float* yb = y + (size_t)b * OUT_CH * TOUT + t0 + ln;

    for (int tt = 0; tt < TILES_PER_WAVE; ++tt) {
        const int td = tt * 16;
        v8f acc = {};
        #pragma unroll
        for (int j = 0; j < NCHUNK; ++j) {
            v2f bm;
            bm[0] = x[offs0[j] + td];
            bm[1] = x[offs1[j] + td];
            // D = A x B + C   (16x16x4 fp32)
            acc = __builtin_amdgcn_wmma_f32_16x16x4_f32(
                false, A[j], false, bm, (short)0, acc, false, false);
        }
        // write-once streaming output -> non-temporal stores
        #pragma unroll
        for (int r = 0; r < 8; ++r) {
            const int o = r + 8 * half;
            __builtin_nontemporal_store(acc[r], yb + (size_t)o * TOUT + td);
        }
    }
}

// ---------------------------------------------------------------------------
extern "C" void kernel_launch(void* const* d_in, const int* in_sizes, int n_in,
                              void* d_out, int out_size, void* d_ws, size_t ws_size,
                              hipStream_t stream) {
    const float* x  = (const float*)d_in[0];
    const float* p  = (const float*)d_in[1];
    const float* W1 = (const float*)d_in[2];
    const float* b1 = (const float*)d_in[3];
    const float* W2 = (const float*)d_in[4];
    const float* b2 = (const float*)d_in[5];
    float* y    = (float*)d_out;
    float* wbuf = (float*)d_ws;   // BATCH * 48 * 16 floats = 96 KB scratch

    hipLaunchKernelGGL(hypernet_kernel, dim3(BATCH), dim3(256), 0, stream,
                       p, W1, b1, W2, b2, wbuf);

    const int total_waves = BATCH * (TOUT / (16 * TILES_PER_WAVE)); // 8192
    hipLaunchKernelGGL(hyperconv_main, dim3(total_waves / 8), dim3(256), 0, stream,
                       x, wbuf, y);
}